// GAT_65979287601498
// MI455X (gfx1250) — compile-verified
//
#include <hip/hip_runtime.h>
#include <math.h>

// Problem constants (fixed by the reference)
#define NN   2000     // nodes
#define NE   32000    // edges
#define EMB  768
#define HID  384
#define FT   768
#define NH   4
#define NEG_SLOPE 0.2f

typedef __attribute__((ext_vector_type(2))) float v2f;
typedef __attribute__((ext_vector_type(8))) float v8f;

static inline int cdiv(long long a, long long b) { return (int)((a + b - 1) / b); }

// ---------------- order-preserving float<->uint for atomicMax softmax ----------------
__device__ __forceinline__ unsigned fenc(float f) {
  unsigned u = __float_as_uint(f);
  return (u & 0x80000000u) ? ~u : (u | 0x80000000u);
}
__device__ __forceinline__ float fdec(unsigned e) {
  unsigned u = (e & 0x80000000u) ? (e ^ 0x80000000u) : ~e;
  return __uint_as_float(u);
}

// ---------------- embedding gathers ----------------
__global__ void k_embed_x(const int* __restrict__ x, const float* __restrict__ emb,
                          float* __restrict__ xf) {
  long long i = (long long)blockIdx.x * blockDim.x + threadIdx.x;
  if (i >= (long long)NN * EMB) return;
  int n = (int)(i / EMB), c = (int)(i % EMB);
  xf[i] = emb[(size_t)x[n * 2 + 0] * EMB + c] + emb[(size_t)x[n * 2 + 1] * EMB + c];
}

__global__ void k_embed_e(const int* __restrict__ ea_idx, const float* __restrict__ emb,
                          float* __restrict__ ea) {
  long long i = (long long)blockIdx.x * blockDim.x + threadIdx.x;
  if (i >= (long long)NE * EMB) return;
  int e = (int)(i / EMB), c = (int)(i % EMB);
  ea[i] = emb[(size_t)ea_idx[e * 2 + 0] * EMB + c] + emb[(size_t)ea_idx[e * 2 + 1] * EMB + c];
}

__global__ void k_deg(const int* __restrict__ dst, float* __restrict__ deg) {
  int e = blockIdx.x * blockDim.x + threadIdx.x;
  if (e < NE) atomicAdd(&deg[dst[e]], 1.0f);
}

// sum incoming edge attrs into self-loop rows (rows NE..NE+NN-1 of ea)
__global__ void k_loopsum(const int* __restrict__ dst, float* __restrict__ ea) {
  long long i = (long long)blockIdx.x * blockDim.x + threadIdx.x;
  if (i >= (long long)NE * EMB) return;
  int e = (int)(i / EMB), c = (int)(i % EMB);
  atomicAdd(&ea[(size_t)(NE + dst[e]) * EMB + c], ea[(size_t)e * EMB + c]);
}

__global__ void k_loopdiv(const float* __restrict__ deg, float* __restrict__ ea) {
  long long i = (long long)blockIdx.x * blockDim.x + threadIdx.x;
  if (i >= (long long)NN * EMB) return;
  int n = (int)(i / EMB);
  ea[(size_t)NE * EMB + i] /= fmaxf(deg[n], 1.0f);
}

// ---------------- fp32 WMMA GEMM: C[M,N] = A[M,K] * B[K,N], row-major ----------------
// Block: 256 threads (8 waves), tile 64(M) x 128(N), K-blocked by 32.
// Waves arranged 2(M) x 4(N), each wave owns a 32x32 tile = 2x2 v_wmma_f32_16x16x4_f32 accums.
// N % 128 == 0 and K % 32 == 0 by construction; only M needs guarding.
__global__ __launch_bounds__(256) void k_gemm_wmma_f32(
    const float* __restrict__ A, const float* __restrict__ B, float* __restrict__ C,
    int M, int N, int K) {
  __shared__ __align__(16) float sA[64][32];
  __shared__ __align__(16) float sBt[128][32];   // B tile stored transposed: sBt[n][k]
  const int tid = threadIdx.x;
  const int wave = tid >> 5, lane = tid & 31;
  const int wm = wave >> 2, wn = wave & 3;       // 2 x 4 wave grid
  const int row16 = lane & 15, khalf = lane >> 4;
  const int bm = blockIdx.y * 64, bn = blockIdx.x * 128;

  v8f acc00 = {}, acc01 = {}, acc10 = {}, acc11 = {};

  for (int k0 = 0; k0 < K; k0 += 32) {
    for (int i = tid; i < 64 * 32; i += 256) {        // A tile
      int r = i >> 5, c = i & 31;
      int gr = bm + r;
      sA[r][c] = (gr < M) ? A[(size_t)gr * K + k0 + c] : 0.0f;
    }
    for (int i = tid; i < 32 * 128; i += 256) {       // B tile (coalesced in n, transposed store)
      int kk = i >> 7, n = i & 127;
      sBt[n][kk] = B[(size_t)(k0 + kk) * N + bn + n];
    }
    __syncthreads();

    const float* a0p = &sA[wm * 32 + row16][0];
    const float* a1p = &sA[wm * 32 + 16 + row16][0];
    const float* b0p = &sBt[wn * 32 + row16][0];
    const float* b1p = &sBt[wn * 32 + 16 + row16][0];
#pragma unroll
    for (int ks = 0; ks < 32; ks += 4) {
      // 16x4 fp32 A fragment: lanes 0-15 hold K={ks,ks+1}, lanes 16-31 K={ks+2,ks+3}
      int kk = ks + khalf * 2;
      v2f a0 = *(const v2f*)(a0p + kk);
      v2f a1 = *(const v2f*)(a1p + kk);
      v2f b0 = *(const v2f*)(b0p + kk);
      v2f b1 = *(const v2f*)(b1p + kk);
      acc00 = __builtin_amdgcn_wmma_f32_16x16x4_f32(false, a0, false, b0, (short)0, acc00, false, false);
      acc01 = __builtin_amdgcn_wmma_f32_16x16x4_f32(false, a0, false, b1, (short)0, acc01, false, false);
      acc10 = __builtin_amdgcn_wmma_f32_16x16x4_f32(false, a1, false, b0, (short)0, acc10, false, false);
      acc11 = __builtin_amdgcn_wmma_f32_16x16x4_f32(false, a1, false, b1, (short)0, acc11, false, false);
    }
    __syncthreads();
  }

  // C/D layout: n = lane%16, m = v + 8*(lane/16)
  int gn0 = bn + wn * 32 + row16;
  int gn1 = gn0 + 16;
  int gm0 = bm + wm * 32 + 8 * khalf;
  int gm1 = gm0 + 16;
#pragma unroll
  for (int v = 0; v < 8; ++v) {
    if (gm0 + v < M) {
      C[(size_t)(gm0 + v) * N + gn0] = acc00[v];
      C[(size_t)(gm0 + v) * N + gn1] = acc01[v];
    }
    if (gm1 + v < M) {
      C[(size_t)(gm1 + v) * N + gn0] = acc10[v];
      C[(size_t)(gm1 + v) * N + gn1] = acc11[v];
    }
  }
}

// ---------------- proj[k,h] = sum_c We[k, h*C+c] * ae[h,c]  (collapses he GEMM) ----------
__global__ void k_proj_e(const float* __restrict__ We, const float* __restrict__ ae,
                         float* __restrict__ proj, int C) {
  int i = blockIdx.x * blockDim.x + threadIdx.x;   // over EMB*NH
  if (i >= EMB * NH) return;
  int k = i / NH, h = i % NH;
  const float* wr = We + (size_t)k * (NH * C) + (size_t)h * C;
  const float* ar = ae + (size_t)h * C;
  float s = 0.f;
  for (int c = 0; c < C; ++c) s += wr[c] * ar[c];
  proj[i] = s;
}

// ---------------- per-(node,head) attention scalars from hx ----------------
__global__ void k_alpha_node(const float* __restrict__ hx, const float* __restrict__ a_src,
                             const float* __restrict__ a_dst, float* __restrict__ ans,
                             float* __restrict__ adn, int C) {
  int i = blockIdx.x * blockDim.x + threadIdx.x;   // over NN*NH
  if (i >= NN * NH) return;
  int n = i / NH, h = i % NH;
  const float* hr = hx + (size_t)n * NH * C + (size_t)h * C;
  const float* sr = a_src + (size_t)h * C;
  const float* dr = a_dst + (size_t)h * C;
  float s = 0.f, t = 0.f;
  for (int c = 0; c < C; ++c) { s += hr[c] * sr[c]; t += hr[c] * dr[c]; }
  ans[i] = s;
  adn[i] = t;
}

// ---------------- one wave32 per edge: alpha_e + combine + leaky_relu + seg-max --------
__global__ void k_edge_alpha(const float* __restrict__ ea, const float* __restrict__ proj,
                             const float* __restrict__ ans, const float* __restrict__ adn,
                             const int* __restrict__ src, const int* __restrict__ dst,
                             float* __restrict__ alpha, unsigned* __restrict__ mx) {
  const int tot = NE + NN;
  int e = blockIdx.x * (blockDim.x >> 5) + (threadIdx.x >> 5);
  int lane = threadIdx.x & 31;
  if (e >= tot) return;
  int s = (e < NE) ? src[e] : (e - NE);
  int d = (e < NE) ? dst[e] : (e - NE);
  const float* row = ea + (size_t)e * EMB;
  const float4* p4 = (const float4*)proj;
  float a0 = 0.f, a1 = 0.f, a2 = 0.f, a3 = 0.f;
  for (int k = lane; k < EMB; k += 32) {
    float v = row[k];
    float4 p = p4[k];
    a0 += v * p.x; a1 += v * p.y; a2 += v * p.z; a3 += v * p.w;
  }
#pragma unroll
  for (int off = 16; off > 0; off >>= 1) {
    a0 += __shfl_xor(a0, off, 32);
    a1 += __shfl_xor(a1, off, 32);
    a2 += __shfl_xor(a2, off, 32);
    a3 += __shfl_xor(a3, off, 32);
  }
  if (lane < 4) {
    float ale = (lane == 0) ? a0 : (lane == 1) ? a1 : (lane == 2) ? a2 : a3;
    float a = ans[s * NH + lane] + adn[d * NH + lane] + ale;
    a = (a > 0.f) ? a : NEG_SLOPE * a;
    alpha[(size_t)e * NH + lane] = a;
    atomicMax(&mx[d * NH + lane], fenc(a));
  }
}

// ---------------- exp(alpha - mx[dst]) and denominator ----------------
__global__ void k_softmax_den(const int* __restrict__ dst, const unsigned* __restrict__ mx,
                              float* __restrict__ alpha, float* __restrict__ den) {
  const int tot = NE + NN;
  long long i = (long long)blockIdx.x * blockDim.x + threadIdx.x;
  if (i >= (long long)tot * NH) return;
  int e = (int)(i >> 2), h = (int)(i & 3);
  int d = (e < NE) ? dst[e] : (e - NE);
  float ex = expf(alpha[i] - fdec(mx[d * NH + h]));
  alpha[i] = ex;                 // reuse buffer as ex
  atomicAdd(&den[d * NH + h], ex);
}

// ---------------- out[d, c] += (1/H) * sum_h att[e,h] * hx[s, h*C + c] ----------------
__global__ void k_aggregate(const float* __restrict__ hx, const float* __restrict__ ex,
                            const float* __restrict__ den, const int* __restrict__ src,
                            const int* __restrict__ dst, float* __restrict__ out, int C) {
  int e = blockIdx.x;
  int s = (e < NE) ? src[e] : (e - NE);
  int d = (e < NE) ? dst[e] : (e - NE);
  __shared__ float att[NH];
  if (threadIdx.x < NH)
    att[threadIdx.x] = ex[(size_t)e * NH + threadIdx.x] / den[d * NH + threadIdx.x];
  __syncthreads();
  const float* hr = hx + (size_t)s * NH * C;
  for (int c = threadIdx.x; c < C; c += blockDim.x) {
    float v = 0.25f * (att[0] * hr[c] + att[1] * hr[C + c] +
                       att[2] * hr[2 * C + c] + att[3] * hr[3 * C + c]);
    atomicAdd(&out[(size_t)d * C + c], v);
  }
}

__global__ void k_bias_relu(const float* __restrict__ in, const float* __restrict__ b,
                            float* __restrict__ o) {
  long long i = (long long)blockIdx.x * blockDim.x + threadIdx.x;
  if (i >= (long long)NN * HID) return;
  o[i] = fmaxf(in[i] + b[i % HID], 0.f);
}

__global__ void k_init_out(const float* __restrict__ b, float* __restrict__ o) {
  long long i = (long long)blockIdx.x * blockDim.x + threadIdx.x;
  if (i >= (long long)NN * FT) return;
  o[i] = b[i % FT];
}

// =====================================================================================
extern "C" void kernel_launch(void* const* d_in, const int* in_sizes, int n_in,
                              void* d_out, int out_size, void* d_ws, size_t ws_size,
                              hipStream_t stream) {
  const int* x      = (const int*)d_in[0];
  const int* eidx   = (const int*)d_in[1];
  const int* src    = eidx;
  const int* dst    = eidx + NE;
  const int* eattr  = (const int*)d_in[2];
  const float* xemb = (const float*)d_in[3];
  const float* eemb = (const float*)d_in[4];
  const float* W1   = (const float*)d_in[5];
  const float* as1  = (const float*)d_in[6];
  const float* ad1  = (const float*)d_in[7];
  const float* We1  = (const float*)d_in[8];
  const float* ae1  = (const float*)d_in[9];
  const float* b1   = (const float*)d_in[10];
  const float* W2   = (const float*)d_in[11];
  const float* as2  = (const float*)d_in[12];
  const float* ad2  = (const float*)d_in[13];
  const float* We2  = (const float*)d_in[14];
  const float* ae2  = (const float*)d_in[15];
  const float* b2   = (const float*)d_in[16];
  float* out = (float*)d_out;

  // ---- workspace layout (floats), all region sizes multiples of 64 ----
  float* ws = (float*)d_ws;
  size_t off = 0;
  float* EA   = ws + off; off += (size_t)(NE + NN) * EMB;   // edge attrs + self-loop attrs
  float* XF   = ws + off; off += (size_t)NN * EMB;
  float* DEG  = ws + off; off += 2048;
  float* HX1  = ws + off; off += (size_t)NN * NH * HID;
  float* HX2  = ws + off; off += (size_t)NN * NH * FT;
  float* PROJ = ws + off; off += (size_t)EMB * NH;
  float* ANS  = ws + off; off += (size_t)NN * NH;
  float* ADN  = ws + off; off += (size_t)NN * NH;
  float* ALPH = ws + off; off += (size_t)(NE + NN) * NH;
  unsigned* MX = (unsigned*)(ws + off); off += (size_t)NN * NH;
  float* DEN  = ws + off; off += (size_t)NN * NH;
  float* OUT1 = ws + off; off += (size_t)NN * HID;
  float* HB   = ws + off; off += (size_t)NN * HID;
  (void)ws_size; (void)in_sizes; (void)n_in; (void)out_size;

  const int T = 256;

  // ---- zero accumulators (every call: graph replays must be deterministic) ----
  hipMemsetAsync(DEG, 0, 2048 * sizeof(float), stream);
  hipMemsetAsync(EA + (size_t)NE * EMB, 0, (size_t)NN * EMB * sizeof(float), stream);
  hipMemsetAsync(OUT1, 0, (size_t)NN * HID * sizeof(float), stream);
  hipMemsetAsync(MX, 0, (size_t)NN * NH * sizeof(unsigned), stream);
  hipMemsetAsync(DEN, 0, (size_t)NN * NH * sizeof(float), stream);

  // ---- shared preprocessing ----
  k_embed_x<<<cdiv((long long)NN * EMB, T), T, 0, stream>>>(x, xemb, XF);
  k_embed_e<<<cdiv((long long)NE * EMB, T), T, 0, stream>>>(eattr, eemb, EA);
  k_deg<<<cdiv(NE, T), T, 0, stream>>>(dst, DEG);
  k_loopsum<<<cdiv((long long)NE * EMB, T), T, 0, stream>>>(dst, EA);
  k_loopdiv<<<cdiv((long long)NN * EMB, T), T, 0, stream>>>(DEG, EA);

  // ================= layer 1 =================
  {
    dim3 g(NH * HID / 128, cdiv(NN, 64));
    k_gemm_wmma_f32<<<g, 256, 0, stream>>>(XF, W1, HX1, NN, NH * HID, EMB);
  }
  k_proj_e<<<cdiv(EMB * NH, T), T, 0, stream>>>(We1, ae1, PROJ, HID);
  k_alpha_node<<<cdiv(NN * NH, T), T, 0, stream>>>(HX1, as1, ad1, ANS, ADN, HID);
  k_edge_alpha<<<cdiv(NE + NN, 8), 256, 0, stream>>>(EA, PROJ, ANS, ADN, src, dst, ALPH, MX);
  k_softmax_den<<<cdiv((long long)(NE + NN) * NH, T), T, 0, stream>>>(dst, MX, ALPH, DEN);
  k_aggregate<<<NE + NN, HID, 0, stream>>>(HX1, ALPH, DEN, src, dst, OUT1, HID);
  k_bias_relu<<<cdiv((long long)NN * HID, T), T, 0, stream>>>(OUT1, b1, HB);

  // re-zero softmax accumulators for layer 2 (stream-ordered)
  hipMemsetAsync(MX, 0, (size_t)NN * NH * sizeof(unsigned), stream);
  hipMemsetAsync(DEN, 0, (size_t)NN * NH * sizeof(float), stream);

  // ================= layer 2 =================
  {
    dim3 g(NH * FT / 128, cdiv(NN, 64));
    k_gemm_wmma_f32<<<g, 256, 0, stream>>>(HB, W2, HX2, NN, NH * FT, HID);
  }
  k_proj_e<<<cdiv(EMB * NH, T), T, 0, stream>>>(We2, ae2, PROJ, FT);
  k_alpha_node<<<cdiv(NN * NH, T), T, 0, stream>>>(HX2, as2, ad2, ANS, ADN, FT);
  k_edge_alpha<<<cdiv(NE + NN, 8), 256, 0, stream>>>(EA, PROJ, ANS, ADN, src, dst, ALPH, MX);
  k_softmax_den<<<cdiv((long long)(NE + NN) * NH, T), T, 0, stream>>>(dst, MX, ALPH, DEN);
  k_init_out<<<cdiv((long long)NN * FT, T), T, 0, stream>>>(b2, out);
  k_aggregate<<<NE + NN, FT, 0, stream>>>(HX2, ALPH, DEN, src, dst, out, FT);
}